// spatial_smoothing_loss_40029095199366
// MI455X (gfx1250) — compile-verified
//
#include <hip/hip_runtime.h>

// ---------------------------------------------------------------------------
// Spatial smoothing loss for MI455X (gfx1250, wave32).
//   loss = sum_{n,c,f,y,x} ( (stencil_f(X)[n,c,y,x])^2 + 1e-12 )^0.45 / 8388608
// Stencils: f0 = 2c-l-r, f1 = 2c-u-d, f2 = 4c-(4 corners); zero padding.
// ---------------------------------------------------------------------------

typedef __attribute__((ext_vector_type(2))) float v2f;
typedef __attribute__((ext_vector_type(8))) float v8f;

#define Wd   512
#define Hd   512
#define RPB  16                    // rows per block (halo amplification 18/16)
#define TPB  128                   // 4 waves; each thread owns 4 px -> 512 px row
#define NIMG 64                    // 32 batch * 2 channels
#define NBLK (NIMG * (Hd / RPB))   // 2048 partials
#define INV_COUNT (1.0f / 8388608.0f)

// t^0.45 = exp2(0.45 * log2(t)) using hardware transcendentals
// (v_log_f32 / v_exp_f32). t = f*f + EPS^2 > 0 always.
__device__ __forceinline__ float pow045(float f) {
    float t = __builtin_fmaf(f, f, 1e-12f);
    return __builtin_amdgcn_exp2f(0.45f * __builtin_amdgcn_logf(t));
}

// Load a 6-wide register window (cols x0-1 .. x0+4) of row y; rows/cols
// outside the image are zero (conv padding = 1).
__device__ __forceinline__ void load_row(const float* __restrict__ img,
                                         int y, int x0, float r[6]) {
    if ((unsigned)y < (unsigned)Hd) {
        const float* p = img + (size_t)y * Wd + x0;
        float4 v = *(const float4*)p;            // aligned b128 load
        r[1] = v.x; r[2] = v.y; r[3] = v.z; r[4] = v.w;
        r[0] = (x0 > 0)       ? p[-1] : 0.0f;    // only thread 0 hits border
        r[5] = (x0 + 4 < Wd)  ? p[4]  : 0.0f;    // only thread 127 hits border
    } else {
        r[0] = r[1] = r[2] = r[3] = r[4] = r[5] = 0.0f;
    }
}

__global__ void __launch_bounds__(TPB)
stencil_partial(const float* __restrict__ X, float* __restrict__ partial) {
    const int x0   = threadIdx.x * 4;     // 4-pixel strip per thread
    const int y0   = blockIdx.x * RPB;    // row strip
    const float* base = X + (size_t)blockIdx.y * (size_t)(Hd * Wd);

    float up[6], mid[6], dn[6];
    load_row(base, y0 - 1, x0, up);
    load_row(base, y0,     x0, mid);

    float acc = 0.0f;
    for (int i = 0; i < RPB; ++i) {
        const int y = y0 + i;
        load_row(base, y + 1, x0, dn);
        if (y + 2 < Hd)   // pull next row toward the caches (global_prefetch_b8)
            __builtin_prefetch(base + (size_t)(y + 2) * Wd + x0, 0, 3);

        #pragma unroll
        for (int j = 0; j < 4; ++j) {
            const float u0 = up[j],  u1 = up[j + 1],  u2 = up[j + 2];
            const float m0 = mid[j], m1 = mid[j + 1], m2 = mid[j + 2];
            const float d0 = dn[j],  d1 = dn[j + 1],  d2 = dn[j + 2];
            const float f0 = 2.0f * m1 - m0 - m2;               // horizontal
            const float f1 = 2.0f * m1 - u1 - d1;               // vertical
            const float f2 = 4.0f * m1 - u0 - u2 - d0 - d2;     // corners
            acc += pow045(f0) + pow045(f1) + pow045(f2);
        }
        #pragma unroll
        for (int k = 0; k < 6; ++k) { up[k] = mid[k]; mid[k] = dn[k]; }
    }

    // Deterministic in-block tree reduction (128 -> 1).
    __shared__ float red[TPB];
    red[threadIdx.x] = acc;
    __syncthreads();
    for (int s = TPB / 2; s > 0; s >>= 1) {
        if (threadIdx.x < s) red[threadIdx.x] += red[threadIdx.x + s];
        __syncthreads();
    }
    if (threadIdx.x == 0)
        partial[blockIdx.y * gridDim.x + blockIdx.x] = red[0];
}

// Final deterministic reduction of 2048 block partials. The last 64->16
// reduction step runs on the matrix unit: D = A x ones, A = 16x4 f32
// (2 VGPRs/lane), so each D row holds a 4-way row sum; lane 0 carries
// row-sums M=0..7 and lane 16 carries M=8..15 in its 8 D VGPRs.
__global__ void __launch_bounds__(256)
final_reduce(const float* __restrict__ partial, float* __restrict__ out, int n) {
    __shared__ float red[256];
    float t = 0.0f;
    for (int i = threadIdx.x; i < n; i += 256) t += partial[i];
    red[threadIdx.x] = t;
    __syncthreads();
    if (threadIdx.x < 128) red[threadIdx.x] += red[threadIdx.x + 128];
    __syncthreads();
    if (threadIdx.x < 64)  red[threadIdx.x] += red[threadIdx.x + 64];
    __syncthreads();

    if (threadIdx.x < 32) {        // wave 0, EXEC all-ones within the wave
        v2f a; a[0] = red[2 * threadIdx.x]; a[1] = red[2 * threadIdx.x + 1];
        v2f b; b[0] = 1.0f; b[1] = 1.0f;
        v8f c = {};
        // 8 args: (neg_a, A, neg_b, B, c_mod, C, reuse_a, reuse_b)
        c = __builtin_amdgcn_wmma_f32_16x16x4_f32(
                false, a, false, b, (short)0, c, false, false);
        float s = c[0] + c[1] + c[2] + c[3] + c[4] + c[5] + c[6] + c[7];
        red[threadIdx.x] = s;      // lanes 0-15: sum(rows 0-7); 16-31: rows 8-15
    }
    __syncthreads();
    if (threadIdx.x == 0)
        out[0] = (red[0] + red[16]) * INV_COUNT;
}

extern "C" void kernel_launch(void* const* d_in, const int* in_sizes, int n_in,
                              void* d_out, int out_size, void* d_ws, size_t ws_size,
                              hipStream_t stream) {
    const float* X  = (const float*)d_in[0];   // (32, 2, 512, 512) f32
    float* out      = (float*)d_out;           // scalar f32
    float* ws       = (float*)d_ws;            // 2048 partials (8 KB)

    dim3 grid(Hd / RPB, NIMG);                 // 32 x 64 = 2048 blocks
    stencil_partial<<<grid, TPB, 0, stream>>>(X, ws);
    final_reduce<<<1, 256, 0, stream>>>(ws, out, NBLK);
}